// OffDiagMatrixConstructionBlock_3547642987269
// MI455X (gfx1250) — compile-verified
//
#include <hip/hip_runtime.h>

typedef float v2f __attribute__((ext_vector_type(2)));
typedef float v8f __attribute__((ext_vector_type(8)));

#define NGc 8
#define Nc  128
#define Cc  128
#define Fc  16
#define Ec  32768
#define NBc 8

#define RC_INV  0.08838834764831845f   // 1/sqrt(128)
#define RNB_INV 0.3535533905932738f    // 1/sqrt(8)
#define R64_INV 0.125f                 // 1/sqrt(64)
#define RK_INV  0.5f                   // 1/sqrt(4)
#define C3      0.5773502691896258f    // 1/sqrt(3)

// LDS pitches (even -> 8B-aligned float2 A-operand loads; odd bank phase)
#define EP   12
#define YP   12
#define SYMP 132
#define ACTP 68
#define WP   388
#define FP   132
#define GP   196

static __device__ __forceinline__ v8f vzero8() {
  v8f z = {0.f, 0.f, 0.f, 0.f, 0.f, 0.f, 0.f, 0.f};
  return z;
}

// One wave computes a 16x16 f32 tile of A(16xK, row-major, pitch lda) @ B(Kx16
// slice of row-major matrix with pitch ldb). CDNA5 V_WMMA_F32_16X16X4_F32:
//   A 16x4: lanes0-15 hold (K=kb+0,kb+1), lanes16-31 hold (K=kb+2,kb+3), M=lane&15
//   B 4x16: lanes0-15 hold rows kb+0/kb+1, lanes16-31 rows kb+2/kb+3, N=lane&15
__device__ __forceinline__ v8f wmma_tile_f32(const float* __restrict__ A, int lda,
                                             const float* __restrict__ B, int ldb,
                                             int K, v8f acc) {
  const int lane = threadIdx.x & 31;
  const int mn = lane & 15;
  const int kh = (lane >> 4) << 1;
#pragma unroll 8
  for (int kb = 0; kb < K; kb += 4) {
    v2f a, b;
    a[0] = A[mn * lda + kb + kh];
    a[1] = A[mn * lda + kb + kh + 1];
    b[0] = B[(kb + kh) * ldb + mn];
    b[1] = B[(kb + kh + 1) * ldb + mn];
    acc = __builtin_amdgcn_wmma_f32_16x16x4_f32(false, a, false, b, (short)0, acc,
                                                false, false);
  }
  return acc;
}

// C/D 16x16 f32 layout: VGPR r holds row (r + 8*(lane>>4)), col = lane&15
__device__ __forceinline__ void tile_store_lds(float* S, int lds, v8f d,
                                               float scale, bool silu) {
  const int lane = threadIdx.x & 31;
  const int n = lane & 15;
  const int rb = (lane >> 4) * 8;
#pragma unroll
  for (int r = 0; r < 8; ++r) {
    float x = d[r] * scale;
    if (silu) x = x * (1.0f / (1.0f + __expf(-x)));
    S[(rb + r) * lds + n] = x;
  }
}

// ---------------- K0: NPW = (node_feats @ W_tp) * C^-1/2  (1024x128) --------
__global__ __launch_bounds__(128) void node_w_kernel(
    const float* __restrict__ node_feats, const float* __restrict__ W_tp,
    float* __restrict__ NPW) {
  const int wv = threadIdx.x >> 5;
  const int tile = blockIdx.x * 4 + wv;   // 512 tiles: 64 row-tiles x 8 col-tiles
  const int row = tile >> 3;
  const int col = tile & 7;
  v8f acc = vzero8();
  acc = wmma_tile_f32(node_feats + (size_t)row * 16 * Cc, Cc, W_tp + col * 16, Cc,
                      Cc, acc);
  const int lane = threadIdx.x & 31;
  const int n = lane & 15;
  const int rb = (lane >> 4) * 8;
#pragma unroll
  for (int r = 0; r < 8; ++r)
    NPW[(size_t)(row * 16 + rb + r) * Cc + col * 16 + n] = acc[r] * RC_INV;
}

// ---------------- zero the dense accumulator --------------------------------
__global__ void zero_kernel(float4* __restrict__ p) {
  size_t i = (size_t)blockIdx.x * blockDim.x + threadIdx.x;
  p[i] = make_float4(0.f, 0.f, 0.f, 0.f);
}

// ---------------- K1: fused per-16-edge pipeline ----------------------------
__global__ __launch_bounds__(128) void edge_kernel(
    const float* __restrict__ node_feats, const float* __restrict__ edge_feats,
    const float* __restrict__ edge_attrs, const int* __restrict__ edge_index,
    const float* __restrict__ W_mid, const float* __restrict__ M1,
    const float* __restrict__ M2, const float* __restrict__ M3,
    const float* __restrict__ M4, const float* __restrict__ P0,
    const float* __restrict__ P1, const float* __restrict__ P2,
    const float* __restrict__ A0, const float* __restrict__ A1,
    const float* __restrict__ A2, const float* __restrict__ NPW,
    float* __restrict__ dense) {
  __shared__ float sE[16 * EP];
  __shared__ float sY[16 * YP];
  __shared__ float sSym[16 * SYMP];
  __shared__ float sH1[16 * ACTP];
  __shared__ float sH2[16 * ACTP];
  __shared__ float sW[16 * WP];
  __shared__ float sF[16 * FP];
  __shared__ float sG[16 * GP];
  __shared__ int sSnd[16];
  __shared__ int sRloc[16];
  __shared__ float sMask[16];

  const int t = threadIdx.x;
  const int wv = t >> 5;
  const int e0 = blockIdx.x * 16;

  if (t == 0) {  // warm weights into near caches (global_prefetch_b8)
    __builtin_prefetch(W_mid, 0, 1);
    __builtin_prefetch(M4, 0, 1);
    __builtin_prefetch(P0, 0, 1);
    __builtin_prefetch(P1, 0, 1);
    __builtin_prefetch(P2, 0, 1);
  }

  // ---- stage 1: gather edge data; sym = xs * (node@W_tp*C^-1/2)[rcv] ----
  {
    const int e = t >> 3;
    const int l8 = t & 7;
    const int ge = e0 + e;
    const int snd = edge_index[ge];
    const int rcv = edge_index[Ec + ge];
    if (l8 == 0) {
      sSnd[e] = snd;
      sRloc[e] = rcv & (Nc - 1);
      sMask[e] = (snd < rcv) ? 1.0f : 0.0f;
    }
    sE[e * EP + l8] = edge_feats[ge * NBc + l8];
    for (int j = l8; j < 9; j += 8) sY[e * YP + j] = edge_attrs[ge * 9 + j];
    const float* xs = node_feats + (size_t)snd * Cc;
    const float* xw = NPW + (size_t)rcv * Cc;
#pragma unroll
    for (int c = 0; c < 16; ++c) {
      const int cc = l8 * 16 + c;
      sSym[e * SYMP + cc] = xs[cc] * xw[cc];
    }
  }
  __syncthreads();

  // ---- MLP1: silu((16x8)@(8x64) * NB^-1/2), wave -> one 16-col tile ----
  {
    v8f acc = vzero8();
    acc = wmma_tile_f32(sE, EP, M1 + wv * 16, 64, NBc, acc);
    tile_store_lds(sH1 + wv * 16, ACTP, acc, RNB_INV, true);
  }
  __syncthreads();
  // ---- MLP2 ----
  {
    v8f acc = vzero8();
    acc = wmma_tile_f32(sH1, ACTP, M2 + wv * 16, 64, 64, acc);
    tile_store_lds(sH2 + wv * 16, ACTP, acc, R64_INV, true);
  }
  __syncthreads();
  // ---- MLP3 ----
  {
    v8f acc = vzero8();
    acc = wmma_tile_f32(sH2, ACTP, M3 + wv * 16, 64, 64, acc);
    tile_store_lds(sH1 + wv * 16, ACTP, acc, R64_INV, true);
  }
  __syncthreads();
  // ---- w = h@M4 * 64^-1/2 : 24 col-tiles, 6 per wave ----
  for (int i = 0; i < 6; ++i) {
    const int ct = wv + 4 * i;
    v8f acc = vzero8();
    acc = wmma_tile_f32(sH1, ACTP, M4 + ct * 16, 3 * Cc, 64, acc);
    tile_store_lds(sW + ct * 16, WP, acc, R64_INV, false);
  }
  // ---- f = sym@W_mid * C^-1/2 : 8 col-tiles, 2 per wave ----
  for (int i = 0; i < 2; ++i) {
    const int ct = wv + 4 * i;
    v8f acc = vzero8();
    acc = wmma_tile_f32(sSym, SYMP, W_mid + ct * 16, Cc, Cc, acc);
    tile_store_lds(sF + ct * 16, FP, acc, RC_INV, false);
  }
  __syncthreads();
  // ---- sW[:, j*128+c] *= f[:, c]  (m_j = f*w_j; Y factors commute out) ----
  {
    const int e = t >> 3;
    const int l8 = t & 7;
    for (int idx = l8; idx < 3 * Cc; idx += 8)
      sW[e * WP + idx] *= sF[e * FP + (idx & (Cc - 1))];
  }
  __syncthreads();
  // ---- g_j = (f*w_j)@P_j * C^-1/2 : 12 tiles, 3 per wave ----
  for (int i = 0; i < 3; ++i) {
    const int tid = wv + 4 * i;
    const int j = tid >> 2;
    const int ct = tid & 3;
    const float* Pm = (j == 0) ? P0 : ((j == 1) ? P1 : P2);
    v8f acc = vzero8();
    acc = wmma_tile_f32(sW + j * Cc, WP, Pm + ct * 16, 64, Cc, acc);
    tile_store_lds(sG + j * 64 + ct * 16, GP, acc, RC_INV, false);
  }
  __syncthreads();

  // ---- assembly: K=4 contractions, 4x4 block H, scatter-add ----
  {
    const int e = t & 15;
    const int f0 = t >> 4;  // handles f0 and f0+8
    const float maskf = sMask[e];
    if (maskf != 0.0f) {
      const int snd = sSnd[e];
      const int rl = sRloc[e];
      float* dbase = dense + ((size_t)snd * Nc + rl) * (Fc * 16);
      const float Y0 = sY[e * YP + 0];
      const float Y1m[3] = {sY[e * YP + 1], sY[e * YP + 2], sY[e * YP + 3]};
      const float Y2m[5] = {sY[e * YP + 4], sY[e * YP + 5], sY[e * YP + 6],
                            sY[e * YP + 7], sY[e * YP + 8]};
#pragma unroll
      for (int ff = 0; ff < 2; ++ff) {
        const int f = f0 + ff * 8;
        float a0 = 0.f, a1 = 0.f, t0 = 0.f, t1 = 0.f, qs = 0.f;
#pragma unroll
        for (int k = 0; k < 4; ++k) {
          const float g0 = sG[e * GP + f * 4 + k];
          const float g1 = sG[e * GP + 64 + f * 4 + k];
          const float g2 = sG[e * GP + 128 + f * 4 + k];
          a0 += g0 * A0[k * 2 + 0];
          a1 += g0 * A0[k * 2 + 1];
          t0 += g1 * A1[k * 2 + 0];
          t1 += g1 * A1[k * 2 + 1];
          qs += g2 * A2[k];
        }
        a0 *= RK_INV * Y0;
        a1 *= RK_INV * Y0;
        t0 *= RK_INV;
        t1 *= RK_INV;
        qs *= RK_INV;
        const float xy = qs * Y2m[0], yz = qs * Y2m[1], z2 = qs * Y2m[2];
        const float xz = qs * Y2m[3], x2 = qs * Y2m[4];
        float H[16];
        H[0] = a0;           H[1] = t0 * Y1m[0];
        H[2] = t0 * Y1m[1];  H[3] = t0 * Y1m[2];
        H[4] = t1 * Y1m[0];  H[5] = a1 * C3 - z2 * C3 + x2;
        H[6] = xy;           H[7] = xz;
        H[8] = t1 * Y1m[1];  H[9] = xy;
        H[10] = a1 * C3 - z2 * C3 - x2;  H[11] = yz;
        H[12] = t1 * Y1m[2]; H[13] = xz;
        H[14] = yz;          H[15] = a1 * C3 + 2.0f * z2 * C3;
        float* dp = dbase + f * 16;
#pragma unroll
        for (int i = 0; i < 16; ++i) atomicAdd(dp + i, H[i]);
      }
    }
  }
}

// ---------------- K2: symmetrize + tile into output -------------------------
__global__ void out_kernel(const float* __restrict__ dense,
                           float* __restrict__ out) {
  const size_t idx = (size_t)blockIdx.x * blockDim.x + threadIdx.x;  // 33.5M
  const int Ccol = (int)(idx & 511);
  const size_t r1 = idx >> 9;
  const int Rrow = (int)(r1 & 511);
  const size_t r2 = r1 >> 9;
  const int f = (int)(r2 & 15);
  const int g = (int)(r2 >> 4);
  const int i = Rrow >> 2, aa = Rrow & 3;
  const int j = Ccol >> 2, bb = Ccol & 3;
  const float v =
      dense[(((size_t)(g * Nc + i) * Nc + j) << 8) + f * 16 + aa * 4 + bb] +
      dense[(((size_t)(g * Nc + j) * Nc + i) << 8) + f * 16 + bb * 4 + aa];
  const size_t o0 = ((((size_t)g * 32 + f) * 512 + Rrow) << 9) + Ccol;
  out[o0] = v;
  out[o0 + (size_t)16 * 512 * 512] = v;  // P=2 tiling
}

extern "C" void kernel_launch(void* const* d_in, const int* in_sizes, int n_in,
                              void* d_out, int out_size, void* d_ws,
                              size_t ws_size, hipStream_t stream) {
  const float* node_feats = (const float*)d_in[0];
  const float* edge_feats = (const float*)d_in[1];
  const float* edge_attrs = (const float*)d_in[2];
  const int* edge_index = (const int*)d_in[3];
  // d_in[4] = batch (derivable from indices; unused)
  const float* W_tp = (const float*)d_in[5];
  const float* W_mid = (const float*)d_in[6];
  const float* M1 = (const float*)d_in[7];
  const float* M2 = (const float*)d_in[8];
  const float* M3 = (const float*)d_in[9];
  const float* M4 = (const float*)d_in[10];
  const float* P0 = (const float*)d_in[11];
  const float* P1 = (const float*)d_in[12];
  const float* P2 = (const float*)d_in[13];
  const float* A0 = (const float*)d_in[14];
  const float* A1 = (const float*)d_in[15];
  const float* A2 = (const float*)d_in[16];

  const size_t dense_elems = (size_t)NGc * Nc * Nc * Fc * 16;  // 33,554,432
  float* dense = (float*)d_ws;
  float* NPW = dense + dense_elems;  // 1024*128 floats
  float* out = (float*)d_out;

  zero_kernel<<<32768, 256, 0, stream>>>((float4*)dense);          // 134 MB -> 0
  node_w_kernel<<<128, 128, 0, stream>>>(node_feats, W_tp, NPW);   // hoisted GEMM
  edge_kernel<<<Ec / 16, 128, 0, stream>>>(
      node_feats, edge_feats, edge_attrs, edge_index, W_mid, M1, M2, M3, M4, P0,
      P1, P2, A0, A1, A2, NPW, dense);
  out_kernel<<<131072, 256, 0, stream>>>(dense, out);
}